// HyperGCN_40587440947831
// MI455X (gfx1250) — compile-verified
//
#include <hip/hip_runtime.h>
#include <stdint.h>

typedef float v2f __attribute__((ext_vector_type(2)));
typedef float v8f __attribute__((ext_vector_type(8)));

#define N_NODES   16384
#define IN_DIM    128
#define HIDDEN    256
#define OUT_DIM   64
#define N_EDGES   8192
#define EDGE_SIZE 8
#define BM_WPR    (N_NODES / 32)          /* 512 bitmap words per row */
#define N_PAIRS   (N_EDGES * EDGE_SIZE * EDGE_SIZE) /* 524288 ordered pairs */

// ---------------------------------------------------------------- zero bitmap
__global__ void hg_zero_kernel(uint4* __restrict__ p, int n4) {
    int t = blockIdx.x * blockDim.x + threadIdx.x;
    if (t < n4) p[t] = make_uint4(0u, 0u, 0u, 0u);
}

// ------------------------------------------------- build dedup'd adjacency bitmap
// A[u,v]=1 for every ordered pair within a hyperedge; plus full diagonal.
// atomicOr gives exact "set to 1" semantics (dedup across edges for free).
__global__ void hg_build_bitmap(const long long* __restrict__ he,
                                uint32_t* __restrict__ bm) {
    int t = blockIdx.x * blockDim.x + threadIdx.x;
    if (t < N_PAIRS) {
        int e = t >> 6;
        int i = (t >> 3) & 7;
        int j = t & 7;
        int u = (int)he[e * EDGE_SIZE + i];
        int v = (int)he[e * EDGE_SIZE + j];
        atomicOr(&bm[(size_t)u * BM_WPR + (v >> 5)], 1u << (v & 31));
    } else {
        int d = t - N_PAIRS;
        if (d < N_NODES)
            atomicOr(&bm[(size_t)d * BM_WPR + (d >> 5)], 1u << (d & 31));
    }
}

// ---------------------------------------------------- degrees -> d^{-1/2}
__global__ void hg_degree_kernel(const uint32_t* __restrict__ bm,
                                 float* __restrict__ dinv) {
    int row = blockIdx.x * blockDim.x + threadIdx.x;
    if (row >= N_NODES) return;
    const uint32_t* wr = bm + (size_t)row * BM_WPR;
    uint32_t cnt = 0;
    for (int w = 0; w < BM_WPR; ++w) cnt += (uint32_t)__popc(wr[w]);
    if (cnt < 1u) cnt = 1u;               // clip (diagonal makes this moot)
    dinv[row] = rsqrtf((float)cnt);
}

// --------------------------------------------------------- fp32 WMMA GEMM + bias
// C[M,N] = A[M,K] @ B[K,N] + bias[N], row-major, via V_WMMA_F32_16X16X4_F32.
// Wave tile: MT(M) x 64(N); MT=32 -> 8 WMMAs per k-step sharing 4 B-frags.
// Software-pipelined: fragments for k+4 are fetched before the k WMMAs issue,
// so loads overlap the matrix pipe instead of draining to loadcnt==0 each step.
// Block: 8 waves stacked in M. grid = (N/64, M/(8*MT)), block = 256.
template <int MT>
__global__ void hg_wmma_gemm_bias(const float* __restrict__ A,
                                  const float* __restrict__ B,
                                  const float* __restrict__ bias,
                                  float* __restrict__ C,
                                  int N, int K) {
    const int lane = threadIdx.x & 31;
    const int wave = threadIdx.x >> 5;
    const int l15  = lane & 15;
    const int half = lane >> 4;                 // 0: K 0,1  | 1: K 2,3 (ISA layout)
    const int m0 = (blockIdx.y * 8 + wave) * MT;
    const int n0 = blockIdx.x * 64;

    v8f c0 = {}, c1 = {}, c2 = {}, c3 = {};
    v8f c4 = {}, c5 = {}, c6 = {}, c7 = {};

    const float* ap0 = A + (size_t)(m0 + l15) * K + half * 2;   // rows m0..m0+15
    const float* ap1 = ap0 + (size_t)16 * K;                    // rows m0+16..m0+31
    const float* bp  = B + (size_t)(half * 2) * N + n0 + l15;

    // ---- prologue: fetch k=0 fragments
    v2f a0c, a1c, b0c, b1c, b2c, b3c;
    a0c.x = ap0[0];  a0c.y = ap0[1];
    if (MT == 32) { a1c.x = ap1[0]; a1c.y = ap1[1]; }
    b0c.x = bp[0];   b0c.y = bp[N];
    b1c.x = bp[16];  b1c.y = bp[N + 16];
    b2c.x = bp[32];  b2c.y = bp[N + 32];
    b3c.x = bp[48];  b3c.y = bp[N + 48];

    for (int k0 = 0; k0 < K; k0 += 4) {
        // rotate current fragments, then prefetch next k-step
        v2f a0 = a0c, a1 = a1c, b0 = b0c, b1 = b1c, b2 = b2c, b3 = b3c;
        if (k0 + 4 < K) {
            ap0 += 4; ap1 += 4; bp += 4 * N;
            a0c.x = ap0[0];  a0c.y = ap0[1];
            if (MT == 32) { a1c.x = ap1[0]; a1c.y = ap1[1]; }
            b0c.x = bp[0];   b0c.y = bp[N];
            b1c.x = bp[16];  b1c.y = bp[N + 16];
            b2c.x = bp[32];  b2c.y = bp[N + 32];
            b3c.x = bp[48];  b3c.y = bp[N + 48];
        }
        c0 = __builtin_amdgcn_wmma_f32_16x16x4_f32(false, a0, false, b0, (short)0, c0, false, false);
        c1 = __builtin_amdgcn_wmma_f32_16x16x4_f32(false, a0, false, b1, (short)0, c1, false, false);
        c2 = __builtin_amdgcn_wmma_f32_16x16x4_f32(false, a0, false, b2, (short)0, c2, false, false);
        c3 = __builtin_amdgcn_wmma_f32_16x16x4_f32(false, a0, false, b3, (short)0, c3, false, false);
        if (MT == 32) {
            c4 = __builtin_amdgcn_wmma_f32_16x16x4_f32(false, a1, false, b0, (short)0, c4, false, false);
            c5 = __builtin_amdgcn_wmma_f32_16x16x4_f32(false, a1, false, b1, (short)0, c5, false, false);
            c6 = __builtin_amdgcn_wmma_f32_16x16x4_f32(false, a1, false, b2, (short)0, c6, false, false);
            c7 = __builtin_amdgcn_wmma_f32_16x16x4_f32(false, a1, false, b3, (short)0, c7, false, false);
        }
    }

    // C layout: VGPR r -> row (tile_m + half*8 + r), col n0 + l15 (+0/16/32/48)
    const float bb0 = bias[n0 + l15];
    const float bb1 = bias[n0 + 16 + l15];
    const float bb2 = bias[n0 + 32 + l15];
    const float bb3 = bias[n0 + 48 + l15];
    {
        const int rbase = m0 + half * 8;
#pragma unroll
        for (int r = 0; r < 8; ++r) {
            float* cp = C + (size_t)(rbase + r) * N + n0 + l15;
            cp[0]  = c0[r] + bb0;
            cp[16] = c1[r] + bb1;
            cp[32] = c2[r] + bb2;
            cp[48] = c3[r] + bb3;
        }
    }
    if (MT == 32) {
        const int rbase = m0 + 16 + half * 8;
#pragma unroll
        for (int r = 0; r < 8; ++r) {
            float* cp = C + (size_t)(rbase + r) * N + n0 + l15;
            cp[0]  = c4[r] + bb0;
            cp[16] = c5[r] + bb1;
            cp[32] = c6[r] + bb2;
            cp[48] = c7[r] + bb3;
        }
    }
}

// --------------------------------------- SpMM: H[u,:] = relu(dinv[u] * sum_v dinv[v]*Z[v,:])
// One wave per row; lane L owns columns L, L+32, ..., L+224 (coalesced gathers).
// Bitmap scan is wave-uniform; 32-MB bitmap + 16-MB Z are L2-resident.
__global__ void hg_spmm_relu(const uint32_t* __restrict__ bm,
                             const float* __restrict__ dinv,
                             const float* __restrict__ Z,
                             float* __restrict__ H) {
    const int lane = threadIdx.x & 31;
    const int wave = threadIdx.x >> 5;
    const int row  = blockIdx.x * 8 + wave;

    float acc[8];
#pragma unroll
    for (int k = 0; k < 8; ++k) acc[k] = 0.f;

    const uint32_t* wr = bm + (size_t)row * BM_WPR;
    for (int w = 0; w < BM_WPR; ++w) {
        uint32_t word = wr[w];
        while (word) {
            int b = __builtin_ctz(word);
            word &= word - 1;
            int v = (w << 5) + b;
            float s = dinv[v];
            const float* zp = Z + (size_t)v * HIDDEN + lane;
#pragma unroll
            for (int k = 0; k < 8; ++k) acc[k] = fmaf(s, zp[32 * k], acc[k]);
        }
    }
    const float du = dinv[row];
    float* hp = H + (size_t)row * HIDDEN + lane;
#pragma unroll
    for (int k = 0; k < 8; ++k) hp[32 * k] = fmaxf(du * acc[k], 0.f);
}

// ---------------------------------------------------------------------- launch
extern "C" void kernel_launch(void* const* d_in, const int* in_sizes, int n_in,
                              void* d_out, int out_size, void* d_ws, size_t ws_size,
                              hipStream_t stream) {
    const float*     x  = (const float*)d_in[0];
    const long long* he = (const long long*)d_in[1];
    const float*     W1 = (const float*)d_in[2];
    const float*     b1 = (const float*)d_in[3];
    const float*     W2 = (const float*)d_in[4];
    const float*     b2 = (const float*)d_in[5];
    const float*     Wh = (const float*)d_in[6];
    const float*     bh = (const float*)d_in[7];
    float*           out = (float*)d_out;

    // Workspace layout (all 64KB-aligned offsets), total ~64.1 MB:
    //   [0, 32MB)       adjacency bitmap (16384 x 512 u32)
    //   [32MB, +64KB)   dinv
    //   [.., +16MB)     bufZ (16384 x 256 f32)
    //   [.., +16MB)     bufH (16384 x 256 f32)
    char* ws = (char*)d_ws;
    uint32_t* bm   = (uint32_t*)(ws);
    float*    dinv = (float*)(ws + (size_t)33554432);
    float*    bufZ = (float*)(ws + (size_t)33619968);
    float*    bufH = (float*)(ws + (size_t)50397184);

    const int bm_words = N_NODES * BM_WPR;          // 8388608
    hg_zero_kernel<<<bm_words / 4 / 256, 256, 0, stream>>>((uint4*)bm, bm_words / 4);

    hg_build_bitmap<<<(N_PAIRS + N_NODES + 255) / 256, 256, 0, stream>>>(he, bm);

    hg_degree_kernel<<<N_NODES / 256, 256, 0, stream>>>(bm, dinv);

    // Z = x @ W1 + b1           (16384x128 @ 128x256), 32-row wave tiles
    hg_wmma_gemm_bias<32><<<dim3(HIDDEN / 64, N_NODES / 256), 256, 0, stream>>>(
        x, W1, b1, bufZ, HIDDEN, IN_DIM);
    // H = relu(Anorm @ Z)
    hg_spmm_relu<<<N_NODES / 8, 256, 0, stream>>>(bm, dinv, bufZ, bufH);

    // Z = H @ W2 + b2           (16384x256 @ 256x256), 32-row wave tiles
    hg_wmma_gemm_bias<32><<<dim3(HIDDEN / 64, N_NODES / 256), 256, 0, stream>>>(
        bufH, W2, b2, bufZ, HIDDEN, HIDDEN);
    // H = relu(Anorm @ Z)
    hg_spmm_relu<<<N_NODES / 8, 256, 0, stream>>>(bm, dinv, bufZ, bufH);

    // out = H @ Wh + bh         (16384x256 @ 256x64), 16-row tiles keep 1024 waves in flight
    hg_wmma_gemm_bias<16><<<dim3(OUT_DIM / 64, N_NODES / 128), 256, 0, stream>>>(
        bufH, Wh, bh, out, OUT_DIM, HIDDEN);
}